// PosteriorDecoder_Qz_79611513799409
// MI455X (gfx1250) — compile-verified
//
#include <hip/hip_runtime.h>
#include <hip/hip_bf16.h>
#include <math.h>

typedef __attribute__((ext_vector_type(2))) float v2f;
typedef __attribute__((ext_vector_type(8))) float v8f;

// Shapes
#define Bb 32
#define Tu 128
#define Tm 128
#define Tz 32
#define Tt 288           // Tu+Tm+Tz
#define Vv 10000
#define Hh 512
#define Ee 128
#define Ss 640           // E+H
#define G3 1536          // 3*H
#define APAD 516         // 512 + 4 pad floats: rotates LDS banks by 4 per row -> conflict-free frag reads

__device__ __forceinline__ v8f wmma_f32(v2f a, v2f b, v8f c) {
  // D = A(16x4 f32) * B(4x16 f32) + C(16x16 f32)
  return __builtin_amdgcn_wmma_f32_16x16x4_f32(
      /*neg_a=*/false, a, /*neg_b=*/false, b,
      /*c_mod=*/(short)0, c, /*reuse_a=*/false, /*reuse_b=*/false);
}

__device__ __forceinline__ float sigmoidf_(float x) { return 1.0f / (1.0f + __expf(-x)); }

// Map a 16-row tile of the virtual enc = [u_hiddens | m_hiddens | pv_z_hidden] (per b, 288 rows)
__device__ __forceinline__ const float* enc_tile_ptr(int tile, const float* u_h, const float* m_h,
                                                     const float* z_h, int* b_out, int* seg_out) {
  int b = tile / 18;
  int tt = (tile % 18) * 16;
  *b_out = b;
  if (tt < Tu)            { *seg_out = 0; return u_h + (size_t)(b * Tu + tt) * Hh; }
  else if (tt < Tu + Tm)  { *seg_out = 1; return m_h + (size_t)(b * Tm + (tt - Tu)) * Hh; }
  else                    { *seg_out = 2; return z_h + (size_t)(b * Tz + (tt - Tu - Tm)) * Hh; }
}

// Cooperative 16x512 tile stage into padded LDS (float4 vectorized)
__device__ __forceinline__ void stage_tile(float* As, const float* src, int tid) {
  for (int idx = tid; idx < 16 * (Hh / 4); idx += 256) {
    int row = idx >> 7;           // /128
    int c4  = (idx & 127) << 2;   // float4 column
    const float4 v = *(const float4*)(src + (size_t)row * Hh + c4);
    *(float4*)(&As[row * APAD + c4]) = v;
  }
}

// ---------------- Kernel 0: hterm[b,g] = h0[b,:] . W_attn[g,0:512] + b_attn[g] ----------------
__global__ void hterm_kernel(const float* __restrict__ last_hidden, const float* __restrict__ W_attn,
                             const float* __restrict__ b_attn, float* __restrict__ hterm) {
  int idx = blockIdx.x * 256 + threadIdx.x;
  if (idx >= Bb * Hh) return;
  int b = idx >> 9, g = idx & (Hh - 1);
  const float* h = last_hidden + b * Hh;       // (1,B,H)
  const float* w = W_attn + (size_t)g * (2 * Hh);
  float a = b_attn[g];
  #pragma unroll 8
  for (int i = 0; i < Hh; ++i) a += h[i] * w[i];
  hterm[idx] = a;
}

// ---------------- Kernel 1: fused energy -> tanh -> dot(v_attn) -> attn logits ----------------
// grid = 576 (one per 16-row tile of B*288 rows), block = 256 (8 waves x 4 col-tiles = 512 cols)
__global__ void attn_energy_kernel(const float* __restrict__ u_h, const float* __restrict__ m_h,
                                   const float* __restrict__ z_h, const float* __restrict__ W_attn,
                                   const float* __restrict__ v_attn, const float* __restrict__ hterm,
                                   float* __restrict__ logits) {
  int tile = blockIdx.x;
  int b, seg;
  const float* src = enc_tile_ptr(tile, u_h, m_h, z_h, &b, &seg);
  (void)seg;

  __shared__ float As[16 * APAD];
  __shared__ float rowsum[16];
  int tid = threadIdx.x;
  if (tid < 16) rowsum[tid] = 0.0f;
  stage_tile(As, src, tid);
  __syncthreads();

  int wave = tid >> 5, lane = tid & 31;
  int half = lane >> 4, l15 = lane & 15;
  const float* aBase = &As[l15 * APAD + 2 * half];

  for (int ci = 0; ci < 4; ++ci) {
    int col0 = (wave * 4 + ci) * 16;
    v8f acc = {};
    // B[k=h][n=g] = W_attn[g, 512+h]  (contiguous in h -> float2 loads)
    const float* bBase = W_attn + (size_t)(col0 + l15) * (2 * Hh) + Hh + 2 * half;
    #pragma unroll 4
    for (int k = 0; k < Hh; k += 4) {
      v2f a  = *(const v2f*)(aBase + k);     // ds_load_b64, bank-conflict-free (APAD)
      v2f bf = *(const v2f*)(bBase + k);
      acc = wmma_f32(a, bf, acc);
    }
    int col = col0 + l15;
    float ht = hterm[b * Hh + col];   // includes b_attn
    float va = v_attn[col];
    #pragma unroll
    for (int i = 0; i < 8; ++i) {
      float e = tanhf(acc[i] + ht) * va;
      // sum over the 16 lanes of this half (shfl_xor with masks <16 stays in-half on wave32)
      #pragma unroll
      for (int m = 1; m < 16; m <<= 1) e += __shfl_xor(e, m, 32);
      if (l15 == 0) atomicAdd(&rowsum[i + 8 * half], e);
    }
  }
  __syncthreads();
  if (tid < 16) logits[tile * 16 + tid] = rowsum[tid];
}

// ---------------- Kernel 2: softmax over 288 + context[b,h] ----------------
__global__ void attn_ctx_kernel(const float* __restrict__ logits, const float* __restrict__ u_h,
                                const float* __restrict__ m_h, const float* __restrict__ z_h,
                                float* __restrict__ context) {
  int b = blockIdx.x, tid = threadIdx.x;
  __shared__ float w[Tt];
  __shared__ float red[256];
  float mx = -1e30f;
  for (int t = tid; t < Tt; t += 256) { float v = logits[b * Tt + t]; w[t] = v; mx = fmaxf(mx, v); }
  red[tid] = mx; __syncthreads();
  for (int s = 128; s > 0; s >>= 1) { if (tid < s) red[tid] = fmaxf(red[tid], red[tid + s]); __syncthreads(); }
  mx = red[0]; __syncthreads();
  float sm = 0.0f;
  for (int t = tid; t < Tt; t += 256) { float e = __expf(w[t] - mx); w[t] = e; sm += e; }
  red[tid] = sm; __syncthreads();
  for (int s = 128; s > 0; s >>= 1) { if (tid < s) red[tid] += red[tid + s]; __syncthreads(); }
  float inv = 1.0f / red[0]; __syncthreads();
  for (int h = tid; h < Hh; h += 256) {
    float acc = 0.0f;
    for (int t = 0; t < Tu; ++t) acc += w[t]       * u_h[(size_t)(b * Tu + t) * Hh + h];
    for (int t = 0; t < Tm; ++t) acc += w[Tu + t]  * m_h[(size_t)(b * Tm + t) * Hh + h];
    for (int t = 0; t < Tz; ++t) acc += w[256 + t] * z_h[(size_t)(b * Tz + t) * Hh + h];
    context[b * Hh + h] = acc * inv;
  }
}

// ---------------- Kernel 3: GRU cell; writes st (ws) and hidden outputs (d_out) ----------------
__global__ void gru_kernel(const float* __restrict__ emb_zt, const float* __restrict__ last_hidden,
                           const float* __restrict__ context, const float* __restrict__ W_ih,
                           const float* __restrict__ W_hh, const float* __restrict__ b_ih,
                           const float* __restrict__ b_hh, float* __restrict__ st,
                           float* __restrict__ d_out) {
  int b = blockIdx.x, tid = threadIdx.x;
  __shared__ float gi[G3], gh[G3];
  __shared__ float xs[Ss], hs[Hh];
  for (int s = tid; s < Ss; s += 256) xs[s] = (s < Ee) ? emb_zt[b * Ee + s] : context[b * Hh + (s - Ee)];
  for (int h = tid; h < Hh; h += 256) hs[h] = last_hidden[b * Hh + h];
  __syncthreads();
  for (int j = tid; j < G3; j += 256) {
    const float* wi = W_ih + (size_t)j * Ss;
    const float* wh = W_hh + (size_t)j * Hh;
    float a = b_ih[j], c = b_hh[j];
    #pragma unroll 8
    for (int s = 0; s < Ss; ++s) a += wi[s] * xs[s];
    #pragma unroll 8
    for (int h = 0; h < Hh; ++h) c += wh[h] * hs[h];
    gi[j] = a; gh[j] = c;
  }
  __syncthreads();
  for (int j = tid; j < Hh; j += 256) {
    float r = sigmoidf_(gi[j] + gh[j]);
    float z = sigmoidf_(gi[Hh + j] + gh[Hh + j]);
    float n = tanhf(gi[2 * Hh + j] + r * gh[2 * Hh + j]);
    float hn = (1.0f - z) * n + z * hs[j];
    st[b * Ss + j] = hn;
    d_out[(size_t)Bb * Vv + b * Hh + j] = hn;                      // last_hidden_new
    d_out[(size_t)Bb * Vv + (size_t)Bb * Hh + b * Hh + j] = hn;    // gru_out
  }
  for (int s = tid; s < Ee; s += 256) st[b * Ss + Hh + s] = emb_zt[b * Ee + s];
}

// ---------------- Kernel 4: score_g = st @ W_gen^T + b_gen  (WMMA, 2x625 tiles) ----------------
__global__ void score_gen_kernel(const float* __restrict__ st, const float* __restrict__ W_gen,
                                 const float* __restrict__ b_gen, float* __restrict__ score_g) {
  int tile = blockIdx.x;            // 0..1249
  int rt = tile / 625;              // row tile (b block)
  int ct = tile % 625;              // vocab col tile
  int lane = threadIdx.x & 31;
  int half = lane >> 4, l15 = lane & 15;
  const float* aBase = st    + (size_t)(rt * 16 + l15) * Ss + 2 * half;  // st tile lives in L1/L2
  const float* bBase = W_gen + (size_t)(ct * 16 + l15) * Ss + 2 * half;  // B[k=s][n=v]=W_gen[v,s]
  v8f acc = {};
  #pragma unroll 4
  for (int k = 0; k < Ss; k += 4) {
    v2f a  = *(const v2f*)(aBase + k);
    v2f bf = *(const v2f*)(bBase + k);
    acc = wmma_f32(a, bf, acc);
  }
  int col = ct * 16 + l15;
  float bg = b_gen[col];
  #pragma unroll
  for (int i = 0; i < 8; ++i) {
    int row = rt * 16 + i + 8 * half;
    score_g[(size_t)row * Vv + col] = acc[i] + bg;
  }
}

// ---------------- Kernel 5: copy scores (fused WMMA proj -> tanh -> dot st) ----------------
// grid = 576 tiles (same enc tiling), block = 256: 8 waves x 5 col-tiles = 640 S-cols
__global__ void copy_score_kernel(const float* __restrict__ u_h, const float* __restrict__ m_h,
                                  const float* __restrict__ z_h,
                                  const float* __restrict__ W_cpu, const float* __restrict__ b_cpu,
                                  const float* __restrict__ W_cpm, const float* __restrict__ b_cpm,
                                  const float* __restrict__ W_cppz, const float* __restrict__ b_cppz,
                                  const float* __restrict__ st, float* __restrict__ sc) {
  int tile = blockIdx.x;
  int b, seg;
  const float* src = enc_tile_ptr(tile, u_h, m_h, z_h, &b, &seg);
  const float* W    = (seg == 0) ? W_cpu : (seg == 1) ? W_cpm : W_cppz;
  const float* bias = (seg == 0) ? b_cpu : (seg == 1) ? b_cpm : b_cppz;

  __shared__ float As[16 * APAD];
  __shared__ float rowsum[16];
  int tid = threadIdx.x;
  if (tid < 16) rowsum[tid] = 0.0f;
  stage_tile(As, src, tid);
  __syncthreads();

  int wave = tid >> 5, lane = tid & 31;
  int half = lane >> 4, l15 = lane & 15;
  const float* aBase = &As[l15 * APAD + 2 * half];

  for (int ci = 0; ci < 5; ++ci) {
    int col0 = (wave * 5 + ci) * 16;
    v8f acc = {};
    const float* bBase = W + (size_t)(col0 + l15) * Hh + 2 * half;  // B[k=h][n=s]=W[s,h]
    #pragma unroll 4
    for (int k = 0; k < Hh; k += 4) {
      v2f a  = *(const v2f*)(aBase + k);
      v2f bf = *(const v2f*)(bBase + k);
      acc = wmma_f32(a, bf, acc);
    }
    int scol = col0 + l15;
    float bi = bias[scol];
    float sv = st[b * Ss + scol];
    #pragma unroll
    for (int i = 0; i < 8; ++i) {
      float e = tanhf(acc[i] + bi) * sv;
      #pragma unroll
      for (int m = 1; m < 16; m <<= 1) e += __shfl_xor(e, m, 32);
      if (l15 == 0) atomicAdd(&rowsum[i + 8 * half], e);
    }
  }
  __syncthreads();
  if (tid < 16) sc[tile * 16 + tid] = rowsum[tid];
}

// ---------------- Kernel 6: 10288-way softmax + p_pz gather + index scatter ----------------
__global__ void final_kernel(const float* __restrict__ score_g, const float* __restrict__ sc,
                             const float* __restrict__ pv_z_prob, const int* __restrict__ u_input,
                             const int* __restrict__ m_input, float* __restrict__ prob_out) {
  int b = blockIdx.x, tid = threadIdx.x;
  __shared__ float red[256];
  __shared__ float pcopy[Tt];
  float mx = -1e30f;
  for (int v = tid; v < Vv; v += 256) mx = fmaxf(mx, score_g[(size_t)b * Vv + v]);
  for (int t = tid; t < Tt; t += 256) mx = fmaxf(mx, sc[b * Tt + t]);
  red[tid] = mx; __syncthreads();
  for (int s = 128; s > 0; s >>= 1) { if (tid < s) red[tid] = fmaxf(red[tid], red[tid + s]); __syncthreads(); }
  mx = red[0]; __syncthreads();
  float sm = 0.0f;
  for (int v = tid; v < Vv; v += 256) sm += __expf(score_g[(size_t)b * Vv + v] - mx);
  for (int t = tid; t < Tt; t += 256) { float e = __expf(sc[b * Tt + t] - mx); pcopy[t] = e; sm += e; }
  red[tid] = sm; __syncthreads();
  for (int s = 128; s > 0; s >>= 1) { if (tid < s) red[tid] += red[tid + s]; __syncthreads(); }
  float inv = 1.0f / red[0]; __syncthreads();

  // p_g + p_pz @ pv_z_prob  (one pass over the 40MB pv_z_prob, the memory floor)
  for (int v = tid; v < Vv; v += 256) {
    float acc = __expf(score_g[(size_t)b * Vv + v] - mx);
    #pragma unroll
    for (int t = 0; t < Tz; ++t) acc += pcopy[256 + t] * pv_z_prob[(size_t)(b * Tz + t) * Vv + v];
    prob_out[(size_t)b * Vv + v] = acc * inv;
  }
  __syncthreads();
  // scatter via indices (replaces 2x164MB one-hot reads)
  for (int t = tid; t < Tu; t += 256) {
    atomicAdd(&prob_out[(size_t)b * Vv + u_input[b * Tu + t]], pcopy[t] * inv);
    atomicAdd(&prob_out[(size_t)b * Vv + m_input[b * Tm + t]], pcopy[Tu + t] * inv);
  }
}

extern "C" void kernel_launch(void* const* d_in, const int* in_sizes, int n_in,
                              void* d_out, int out_size, void* d_ws, size_t ws_size,
                              hipStream_t stream) {
  const int*   u_input  = (const int*)  d_in[0];
  const float* u_h      = (const float*)d_in[2];
  const int*   m_input  = (const int*)  d_in[3];
  const float* m_h      = (const float*)d_in[5];
  const float* pv_zp    = (const float*)d_in[6];
  const float* z_h      = (const float*)d_in[7];
  const float* emb_zt   = (const float*)d_in[9];
  const float* last_h   = (const float*)d_in[10];
  const float* W_attn   = (const float*)d_in[11];
  const float* b_attn   = (const float*)d_in[12];
  const float* v_attn   = (const float*)d_in[13];
  const float* W_ih     = (const float*)d_in[14];
  const float* W_hh     = (const float*)d_in[15];
  const float* b_ih     = (const float*)d_in[16];
  const float* b_hh     = (const float*)d_in[17];
  const float* W_gen    = (const float*)d_in[18];
  const float* b_gen    = (const float*)d_in[19];
  const float* W_cpu    = (const float*)d_in[20];
  const float* b_cpu    = (const float*)d_in[21];
  const float* W_cpm    = (const float*)d_in[22];
  const float* b_cpm    = (const float*)d_in[23];
  const float* W_cppz   = (const float*)d_in[24];
  const float* b_cppz   = (const float*)d_in[25];
  float* out = (float*)d_out;

  // workspace layout (floats)
  float* ws       = (float*)d_ws;
  float* hterm    = ws;                          // 32*512   = 16384
  float* alogits  = hterm   + Bb * Hh;           // 32*288   =  9216
  float* context  = alogits + Bb * Tt;           // 32*512   = 16384
  float* st       = context + Bb * Hh;           // 32*640   = 20480
  float* score_g  = st      + Bb * Ss;           // 32*10000 = 320000
  float* sc       = score_g + (size_t)Bb * Vv;   // 32*288   =  9216

  hipLaunchKernelGGL(hterm_kernel, dim3(64), dim3(256), 0, stream, last_h, W_attn, b_attn, hterm);
  hipLaunchKernelGGL(attn_energy_kernel, dim3((Bb * Tt) / 16), dim3(256), 0, stream,
                     u_h, m_h, z_h, W_attn, v_attn, hterm, alogits);
  hipLaunchKernelGGL(attn_ctx_kernel, dim3(Bb), dim3(256), 0, stream, alogits, u_h, m_h, z_h, context);
  hipLaunchKernelGGL(gru_kernel, dim3(Bb), dim3(256), 0, stream,
                     emb_zt, last_h, context, W_ih, W_hh, b_ih, b_hh, st, out);
  hipLaunchKernelGGL(score_gen_kernel, dim3(2 * (Vv / 16)), dim3(32), 0, stream, st, W_gen, b_gen, score_g);
  hipLaunchKernelGGL(copy_score_kernel, dim3((Bb * Tt) / 16), dim3(256), 0, stream,
                     u_h, m_h, z_h, W_cpu, b_cpu, W_cpm, b_cpm, W_cppz, b_cppz, st, sc);
  hipLaunchKernelGGL(final_kernel, dim3(Bb), dim3(256), 0, stream, score_g, sc, pv_zp, u_input, m_input, out);
}